// GraphEncoderNetwork_8014408974587
// MI455X (gfx1250) — compile-verified
//
#include <hip/hip_runtime.h>
#include <hip/hip_bf16.h>

typedef __attribute__((ext_vector_type(2))) float v2f;
typedef __attribute__((ext_vector_type(8))) float v8f;

#define WPB 8  // waves per block (256 threads, wave32)

// LDS ops from one wave complete in order (DScnt is in-order per the ISA);
// we only need to stop the compiler from reordering across this point.
__device__ __forceinline__ void lds_seq_fence() { asm volatile("" ::: "memory"); }

// -----------------------------------------------------------------------------
// One fully-connected layer on a 16-row tile using V_WMMA_F32_16X16X4_F32
// (full fp32; K stepped 4 at a time).
//   in_tile : [16][KIN]  row-major (LDS)
//   W       : [KIN][NOUT] row-major (global)
//   Bv      : [NOUT] bias (global)
//   out_tile: [16][NOUT] row-major (LDS)
// A-frag: lane<16 -> (A[m][k0],A[m][k0+1]) m=lane; lane>=16 -> K pair {k0+2,k0+3}
// B-frag: lane<16 -> (B[k0][n],B[k0+1][n]) n=lane; lane>=16 -> rows {k0+2,k0+3}
// D: vgpr r, lane l -> D[8*(l>=16)+r][l&15]
// -----------------------------------------------------------------------------
template<int KIN, int NOUT, bool RELU>
__device__ __forceinline__ void wave_linear(const float* in_tile,
                                            const float* __restrict__ W,
                                            const float* __restrict__ Bv,
                                            float* out_tile, int lane)
{
    const int mrow = lane & 15;
    const int hi   = lane >> 4;   // 0: K pair {0,1}; 1: K pair {2,3}
    const int ksel = hi * 2;
#pragma unroll
    for (int nt = 0; nt < NOUT; nt += 16) {
        const int  n      = nt + mrow;
        const bool nvalid = (n < NOUT);   // handles NOUT==8 tail columns
        v8f acc = {};
#pragma unroll
        for (int k0 = 0; k0 < KIN; k0 += 4) {
            v2f a, b;
            a.x = in_tile[mrow * KIN + k0 + ksel];
            a.y = in_tile[mrow * KIN + k0 + ksel + 1];
            b.x = nvalid ? W[(k0 + ksel) * NOUT + n] : 0.0f;
            b.y = nvalid ? W[(k0 + ksel + 1) * NOUT + n] : 0.0f;
            acc = __builtin_amdgcn_wmma_f32_16x16x4_f32(
                    false, a, false, b, (short)0, acc, false, false);
        }
        if (nvalid) {
            const float bias = Bv[n];
#pragma unroll
            for (int r = 0; r < 8; ++r) {
                float v = acc[r] + bias;
                if (RELU) v = fmaxf(v, 0.0f);
                out_tile[(hi * 8 + r) * NOUT + n] = v;
            }
        }
    }
    lds_seq_fence();
}

// 3-layer MLP: D0 -> 32 (relu) -> 16 (relu) -> D3. One 16-row tile per wave.
template<int D0, int D3>
__global__ void __launch_bounds__(256)
mlp3_wmma_kernel(const float* __restrict__ in, float* __restrict__ out,
                 const float* __restrict__ w1, const float* __restrict__ b1,
                 const float* __restrict__ w2, const float* __restrict__ b2,
                 const float* __restrict__ w3, const float* __restrict__ b3,
                 int nrows)
{
    __shared__ float lds[WPB][2][16 * 32];
    const int lane = threadIdx.x & 31;
    const int wid  = threadIdx.x >> 5;
    const int row0 = (blockIdx.x * WPB + wid) * 16;
    if (row0 >= nrows) return;                 // wave-uniform exit
    float* buf0 = &lds[wid][0][0];
    float* buf1 = &lds[wid][1][0];

    // Stage input tile (zero-pad rows past nrows so WMMA runs with full tiles).
    for (int i = lane; i < 16 * D0; i += 32) {
        const int m = i / D0, k = i - m * D0;
        const int row = row0 + m;
        buf0[i] = (row < nrows) ? in[(size_t)row * D0 + k] : 0.0f;
    }
    lds_seq_fence();

    wave_linear<D0, 32, true >(buf0, w1, b1, buf1, lane);
    wave_linear<32, 16, true >(buf1, w2, b2, buf0, lane);
    wave_linear<16, D3, false>(buf0, w3, b3, buf1, lane);

    for (int i = lane; i < 16 * D3; i += 32) {
        const int m = i / D3, k = i - m * D3;
        const int row = row0 + m;
        if (row < nrows) out[(size_t)row * D3 + k] = buf1[i];
    }
}

// deg starts at 1.0 (self loop); y_pool zeroed each call (harness poisons ws).
__global__ void init_kernel(float* __restrict__ deg, float* __restrict__ ypool,
                            int N, int G16)
{
    const int i = blockIdx.x * blockDim.x + threadIdx.x;
    if (i < N)   deg[i] = 1.0f;
    if (i < G16) ypool[i] = 0.0f;
}

__global__ void degree_kernel(const int* __restrict__ col, float* __restrict__ deg, int E)
{
    const int e = blockIdx.x * blockDim.x + threadIdx.x;
    if (e < E) atomicAdd(&deg[col[e]], 1.0f);
}

// deg -> dinv in place; agg = dinv^2 * x1 (the self-loop message, non-atomic).
__global__ void dinv_selfloop_kernel(float* __restrict__ deg_dinv,
                                     const float* __restrict__ x1,
                                     float* __restrict__ agg, int N)
{
    const int i = blockIdx.x * blockDim.x + threadIdx.x;
    if (i >= N) return;
    const float d  = deg_dinv[i];
    const float di = 1.0f / sqrtf(d);   // deg >= 1 always (self loop)
    deg_dinv[i] = di;
    const float w = di * di;
    const float4* xp = (const float4*)(x1 + (size_t)i * 8);
    float4 lo = xp[0], hiv = xp[1];
    float4 olo = {w * lo.x,  w * lo.y,  w * lo.z,  w * lo.w};
    float4 ohi = {w * hiv.x, w * hiv.y, w * hiv.z, w * hiv.w};
    float4* ap = (float4*)(agg + (size_t)i * 8);
    ap[0] = olo; ap[1] = ohi;
}

// One thread per edge: gather x1[col] (L2-resident, 16 MB) and scatter-add
// the normalized message into agg[row].
__global__ void edge_scatter_kernel(const int* __restrict__ row,
                                    const int* __restrict__ col,
                                    const float* __restrict__ dinv,
                                    const float* __restrict__ x1,
                                    float* __restrict__ agg, int E)
{
    const int e = blockIdx.x * blockDim.x + threadIdx.x;
    if (e >= E) return;
    const int r = row[e], c = col[e];
    const float w = dinv[r] * dinv[c];
    const float4* xp = (const float4*)(x1 + (size_t)c * 8);
    const float4 lo = xp[0], hiv = xp[1];
    float* dst = agg + (size_t)r * 8;
    atomicAdd(dst + 0, w * lo.x);  atomicAdd(dst + 1, w * lo.y);
    atomicAdd(dst + 2, w * lo.z);  atomicAdd(dst + 3, w * lo.w);
    atomicAdd(dst + 4, w * hiv.x); atomicAdd(dst + 5, w * hiv.y);
    atomicAdd(dst + 6, w * hiv.z); atomicAdd(dst + 7, w * hiv.w);
}

// global_add_pool: one thread per (node, channel).
__global__ void pool_kernel(const float* __restrict__ xout,
                            const int* __restrict__ batch,
                            float* __restrict__ ypool, int N)
{
    const int idx = blockIdx.x * blockDim.x + threadIdx.x;
    if (idx >= N * 16) return;
    const int i = idx >> 4, c = idx & 15;
    atomicAdd(&ypool[batch[i] * 16 + c], xout[idx]);
}

// z = mlp_global(sum(y, axis=0)); tiny, single block.
__global__ void final_kernel(const float* __restrict__ y,
                             const float* __restrict__ w1, const float* __restrict__ b1,
                             const float* __restrict__ w2, const float* __restrict__ b2,
                             const float* __restrict__ w3, const float* __restrict__ b3,
                             float* __restrict__ z, int G)
{
    __shared__ float ysum[16];
    __shared__ float h1[32];
    __shared__ float h2[16];
    const int t = threadIdx.x;
    if (t < 16) ysum[t] = 0.0f;
    __syncthreads();
    {
        const int c = t & 15;
        float s = 0.0f;
        for (int r = t >> 4; r < G; r += 16) s += y[(size_t)r * 16 + c];
        atomicAdd(&ysum[c], s);
    }
    __syncthreads();
    if (t < 32) {
        float a = b1[t];
        for (int k = 0; k < 16; ++k) a += ysum[k] * w1[k * 32 + t];
        h1[t] = fmaxf(a, 0.0f);
    }
    __syncthreads();
    if (t < 16) {
        float a = b2[t];
        for (int k = 0; k < 32; ++k) a += h1[k] * w2[k * 16 + t];
        h2[t] = fmaxf(a, 0.0f);
    }
    __syncthreads();
    if (t < 16) {
        float a = b3[t];
        for (int k = 0; k < 16; ++k) a += h2[k] * w3[k * 16 + t];
        z[t] = a;
    }
}

extern "C" void kernel_launch(void* const* d_in, const int* in_sizes, int n_in,
                              void* d_out, int out_size, void* d_ws, size_t ws_size,
                              hipStream_t stream)
{
    const float* x     = (const float*)d_in[0];
    const int*   eidx  = (const int*)d_in[1];   // [2, E] flat
    const int*   batch = (const int*)d_in[2];
    const int N = in_sizes[0] / 8;
    const int E = in_sizes[1] / 2;
    const int G = 1000;

    // MLP params: 4 groups of 6 tensors at inputs 3..26. Detect flattening
    // order from sizes: biases are <=32 elems, weights >=128 elems.
    // insertion order: w1,b1,w2,b2,w3,b3 ; jax-tree-sorted: b1,b2,b3,w1,w2,w3.
    const float* P[4][6]; // [mlp][w1,b1,w2,b2,w3,b3]
    for (int g = 0; g < 4; ++g) {
        const int base = 3 + 6 * g;
        const bool bias_first = (in_sizes[base] <= 32);
        if (bias_first) {
            P[g][0] = (const float*)d_in[base + 3]; // w1
            P[g][1] = (const float*)d_in[base + 0]; // b1
            P[g][2] = (const float*)d_in[base + 4]; // w2
            P[g][3] = (const float*)d_in[base + 1]; // b2
            P[g][4] = (const float*)d_in[base + 5]; // w3
            P[g][5] = (const float*)d_in[base + 2]; // b3
        } else {
            for (int j = 0; j < 6; ++j) P[g][j] = (const float*)d_in[base + j];
        }
    }

    // Workspace layout (floats): x1[N*8] | agg[N*8] | deg->dinv[N] | ypool[G*16]
    float* ws    = (float*)d_ws;
    float* x1    = ws;
    float* agg   = ws + (size_t)N * 8;
    float* dinv  = ws + (size_t)N * 16;
    float* ypool = ws + (size_t)N * 17;

    float* xout = (float*)d_out;                  // [N,16]
    float* yout = xout + (size_t)N * 16;          // [G,16]
    float* zout = yout + (size_t)G * 16;          // [16]

    const int* erow = eidx;       // edge_index[0]
    const int* ecol = eidx + E;   // edge_index[1]

    const int tpb = 256;
    const int tilesN = (N + 15) / 16;
    const int tilesG = (G + 15) / 16;

    init_kernel<<<(N + tpb - 1) / tpb, tpb, 0, stream>>>(dinv, ypool, N, G * 16);

    mlp3_wmma_kernel<8, 8><<<(tilesN + WPB - 1) / WPB, tpb, 0, stream>>>(
        x, x1, P[0][0], P[0][1], P[0][2], P[0][3], P[0][4], P[0][5], N);

    degree_kernel<<<(E + tpb - 1) / tpb, tpb, 0, stream>>>(ecol, dinv, E);

    dinv_selfloop_kernel<<<(N + tpb - 1) / tpb, tpb, 0, stream>>>(dinv, x1, agg, N);

    edge_scatter_kernel<<<(E + tpb - 1) / tpb, tpb, 0, stream>>>(
        erow, ecol, dinv, x1, agg, E);

    mlp3_wmma_kernel<8, 16><<<(tilesN + WPB - 1) / WPB, tpb, 0, stream>>>(
        agg, xout, P[1][0], P[1][1], P[1][2], P[1][3], P[1][4], P[1][5], N);

    pool_kernel<<<(N * 16 + tpb - 1) / tpb, tpb, 0, stream>>>(xout, batch, ypool, N);

    mlp3_wmma_kernel<16, 16><<<(tilesG + WPB - 1) / WPB, tpb, 0, stream>>>(
        ypool, yout, P[2][0], P[2][1], P[2][2], P[2][3], P[2][4], P[2][5], G);

    final_kernel<<<1, tpb, 0, stream>>>(
        yout, P[3][0], P[3][1], P[3][2], P[3][3], P[3][4], P[3][5], zout, G);

    (void)n_in; (void)out_size; (void)ws_size;
}